// IcoGenericUpConv_8641474199780
// MI455X (gfx1250) — compile-verified
//
#include <hip/hip_runtime.h>

// Problem constants (from reference)
#define B_DIM  16
#define C_IN   256
#define C_OUTD 128
#define N_LOWV 10242
#define N_UPV  40962
#define KNB    7
#define FDIM   (KNB * C_OUTD)   // 896

typedef __attribute__((ext_vector_type(2))) float v2f;
typedef __attribute__((ext_vector_type(8))) float v8f;
typedef __attribute__((ext_vector_type(4))) unsigned int u32x4;
typedef __attribute__((ext_vector_type(8))) int i32x8;
typedef __attribute__((ext_vector_type(4))) int i32x4;

// ---------------------------------------------------------------------------
// Fused GEMM (fp32 WMMA 16x16x4) + bias + scatter-add epilogue.
//
// Block = 256 threads = 8 waves. The block's 16-vertex x-tile (16 x 256 fp32,
// 16 KB) is DMA'd into LDS once by the Tensor Data Mover (zero-filling any
// rows past N_LOW via tensor_dim0 OOB), then every wave streams A fragments
// from LDS while pulling its private 16 weight columns from L2-resident W.
// Each wave owns a 16(v) x 16(f) tile of Y[b] and atomically scatters rows
// into out[b, c, neigh[v*K+k]].
// ---------------------------------------------------------------------------
__global__ __launch_bounds__(256) void ico_gemm_scatter(
    const float* __restrict__ x,      // [B, C_in, N_low]
    const float* __restrict__ w,      // [C_in, F]
    const float* __restrict__ bias,   // [F]
    const int*   __restrict__ neigh,  // [N_low * K]
    float*       __restrict__ out)    // [B, C_out, N_up] (accumulates sums)
{
    __shared__ float tileA[C_IN * 16];            // [c][v_local], 16 KB

    const int lane = threadIdx.x & 31;
    const int wave = threadIdx.x >> 5;
    const int m    = lane & 15;   // row (A) / col (B,C,D) within tile
    const int half = lane >> 4;   // lane-group: selects K pair / M upper half

    const int v0 = blockIdx.x * 16;                // vertex tile base
    const int f0 = blockIdx.y * 128 + wave * 16;   // f column tile base
    const int b  = blockIdx.z;

    // --- TDM: DMA x[b, 0:256, v0:v0+16] -> LDS (row-major [c][16]) ---------
    if (wave == 0) {
        const float* gptr = x + (size_t)b * C_IN * N_LOWV + v0;
        unsigned long long ga   = (unsigned long long)(uintptr_t)gptr;
        unsigned int       lds0 = (unsigned int)(uintptr_t)(&tileA[0]);
        unsigned int       dim0 = (unsigned int)(N_LOWV - v0); // OOB -> zero fill

        u32x4 g0;
        g0[0] = 1u;                                       // count=1, no gather
        g0[1] = lds0;                                     // lds_addr
        g0[2] = (unsigned int)ga;                         // global_addr[31:0]
        g0[3] = (unsigned int)((ga >> 32) & 0x01FFFFFFu)  // global_addr[56:32]
                | 0x80000000u;                            // type=2 ("image")

        i32x8 g1;
        g1[0] = (int)(2u << 16);                          // data_size=4B; mask=0
        g1[1] = (int)((dim0 & 0xFFFFu) << 16);            // tensor_dim0[15:0]
        g1[2] = (int)((dim0 >> 16) | ((unsigned)C_IN << 16)); // dim0 hi | dim1 lo
        g1[3] = (int)(16u << 16);                         // dim1 hi=0 | tile_dim0=16
        g1[4] = (int)(unsigned)C_IN;                      // tile_dim1=256, tile_dim2=0
        g1[5] = (int)(unsigned)N_LOWV;                    // tensor_dim0_stride lo32
        g1[6] = 0;                                        // stride0 hi | stride1 lo
        g1[7] = 0;                                        // stride1 hi
        i32x4 gz4 = {0, 0, 0, 0};
        i32x8 gz8 = {0, 0, 0, 0, 0, 0, 0, 0};

        __builtin_amdgcn_tensor_load_to_lds(g0, g1, gz4, gz4, gz8, 0);
        __builtin_amdgcn_s_wait_tensorcnt(0);
    }
    __syncthreads();

    // --- WMMA K-loop over C_in in steps of 4 -------------------------------
    // A fragment (16x4 f32): lanes 0-15: M=m, K={c,c+1}; lanes 16-31: K={c+2,c+3}
    // B fragment (4x16 f32): lane%16 = N, same K split across halves.
    const float* aLds  = &tileA[2 * half * 16 + m];      // + c*16
    const float* bLane = w + f0 + m + (size_t)(2 * half) * FDIM; // + c*FDIM

    v8f acc = {0.f, 0.f, 0.f, 0.f, 0.f, 0.f, 0.f, 0.f};

    #pragma unroll 8
    for (int c = 0; c < C_IN; c += 4) {
        v2f a, bm;
        a.x  = aLds[(c + 0) * 16];
        a.y  = aLds[(c + 1) * 16];
        bm.x = bLane[(size_t)(c + 0) * FDIM];
        bm.y = bLane[(size_t)(c + 1) * FDIM];
        acc = __builtin_amdgcn_wmma_f32_16x16x4_f32(
            /*neg_a=*/false, a, /*neg_b=*/false, bm,
            /*c_mod=*/(short)0, acc, /*reuse_a=*/false, /*reuse_b=*/false);
    }

    // --- Epilogue: D VGPR r holds (M = r + 8*half, N = m) ------------------
    // f0 is 16-aligned and C_out=128 => k = f/C_out constant per wave.
    const int   k  = f0 / C_OUTD;
    const int   c0 = f0 % C_OUTD;
    const float bv = bias[f0 + m];
    float* outCol  = out + ((size_t)b * C_OUTD + (c0 + m)) * N_UPV;

    #pragma unroll
    for (int r = 0; r < 8; ++r) {
        const int vrow = v0 + r + 8 * half;
        if (vrow < N_LOWV) {
            const int n = neigh[vrow * KNB + k];   // broadcast within half-wave
            atomicAdd(outCol + n, acc[r] + bv);
        }
    }
}

// counts[n] = number of (v,k) pairs mapping to up-vertex n
__global__ void ico_count_kernel(const int* __restrict__ neigh,
                                 float* __restrict__ counts)
{
    int i = blockIdx.x * blockDim.x + threadIdx.x;
    if (i < N_LOWV * KNB) atomicAdd(&counts[neigh[i]], 1.0f);
}

// counts -> 1/counts (cover permutation in the reference guarantees counts >= 1)
__global__ void ico_invert_kernel(float* __restrict__ counts)
{
    int i = blockIdx.x * blockDim.x + threadIdx.x;
    if (i < N_UPV) counts[i] = 1.0f / counts[i];
}

// out[b,c,n] *= 1/counts[n]   (grid.y enumerates the B*C_out planes)
__global__ void ico_scale_kernel(float* __restrict__ out,
                                 const float* __restrict__ cinv)
{
    int n = blockIdx.x * blockDim.x + threadIdx.x;
    if (n < N_UPV) {
        size_t base = (size_t)blockIdx.y * N_UPV;
        out[base + n] *= cinv[n];
    }
}

extern "C" void kernel_launch(void* const* d_in, const int* in_sizes, int n_in,
                              void* d_out, int out_size, void* d_ws, size_t ws_size,
                              hipStream_t stream)
{
    (void)in_sizes; (void)n_in; (void)ws_size;
    const float* x     = (const float*)d_in[0];
    const float* w     = (const float*)d_in[1];
    const float* bias  = (const float*)d_in[2];
    const int*   neigh = (const int*)d_in[3];
    // d_in[4] is the n_up scalar; dimensions are compile-time constants here.

    float* out    = (float*)d_out;
    float* counts = (float*)d_ws;    // N_UP floats of scratch

    // Zero accumulation buffers (graph-capture-safe async memsets).
    (void)hipMemsetAsync(d_out, 0, (size_t)out_size * sizeof(float), stream);
    (void)hipMemsetAsync(counts, 0, (size_t)N_UPV * sizeof(float), stream);

    // Segment counts and their reciprocals.
    ico_count_kernel<<<(N_LOWV * KNB + 255) / 256, 256, 0, stream>>>(neigh, counts);
    ico_invert_kernel<<<(N_UPV + 255) / 256, 256, 0, stream>>>(counts);

    // Fused WMMA GEMM (TDM-staged A tile) + scatter-add of sums.
    dim3 grid((N_LOWV + 15) / 16, FDIM / 128, B_DIM);
    ico_gemm_scatter<<<grid, 256, 0, stream>>>(x, w, bias, neigh, out);

    // Divide sums by counts -> mean.
    dim3 grid2((N_UPV + 255) / 256, B_DIM * C_OUTD);
    ico_scale_kernel<<<grid2, 256, 0, stream>>>(out, counts);
}